// GCN_23940147708471
// MI455X (gfx1250) — compile-verified
//
#include <hip/hip_runtime.h>
#include <hip/hip_bf16.h>
#include <math.h>

typedef __attribute__((ext_vector_type(2))) float v2f;
typedef __attribute__((ext_vector_type(8))) float v8f;

// ---------------- utility kernels ----------------

__global__ void fill_f32(float* __restrict__ p, float val, long long n) {
  long long i = (long long)blockIdx.x * blockDim.x + threadIdx.x;
  if (i < n) p[i] = val;
}

// deg[col] += 1 per edge (deg pre-initialized to 1.0 for the self loop)
__global__ void deg_count(const long long* __restrict__ cols, float* __restrict__ deg,
                          long long nE) {
  long long e = (long long)blockIdx.x * blockDim.x + threadIdx.x;
  if (e < nE) atomicAdd(&deg[cols[e]], 1.0f);
}

// deg -> rsqrt(deg) in place (deg >= 1 always due to self loops)
__global__ void rsqrt_inplace(float* __restrict__ deg, int n) {
  int i = blockIdx.x * blockDim.x + threadIdx.x;
  if (i < n) deg[i] = rsqrtf(deg[i]);
}

// ---------------- layer 1 GEMM: h1 = x @ W1  ([N,128]x[128,16]) ----------------
// One wave per 16-row tile, fp32 WMMA 16x16x4, 32 MACC steps over K=128.
__global__ void __launch_bounds__(256) gemm1_wmma(const float* __restrict__ x,
                                                  const float* __restrict__ W1,
                                                  float* __restrict__ h1, int n_nodes) {
  __shared__ float sW[128 * 16];
  for (int i = threadIdx.x; i < 128 * 16; i += 256) sW[i] = W1[i];
  __syncthreads();

  const int lane = threadIdx.x & 31;
  const int tile = blockIdx.x * 8 + (threadIdx.x >> 5);
  const int row0 = tile * 16;
  if (row0 >= n_nodes) return;  // whole-wave uniform: EXEC all-1s for WMMA

  const int m = lane & 15;            // M for A-frag, N for B/D frags
  const int koff = (lane >> 4) << 1;  // upper half-wave holds K+2,K+3

  int rA = row0 + m;
  if (rA > n_nodes - 1) rA = n_nodes - 1;  // clamp (N is a multiple of 16 anyway)
  const float* __restrict__ xrow = x + (size_t)rA * 128;

  v8f c = {0.f, 0.f, 0.f, 0.f, 0.f, 0.f, 0.f, 0.f};
#pragma unroll
  for (int k = 0; k < 128; k += 4) {
    v2f a, b;
    a[0] = xrow[k + koff];
    a[1] = xrow[k + koff + 1];
    b[0] = sW[(k + koff) * 16 + m];
    b[1] = sW[(k + koff + 1) * 16 + m];
    c = __builtin_amdgcn_wmma_f32_16x16x4_f32(false, a, false, b, (short)0, c,
                                              false, false);
  }

  // D layout: lanes 0-15 -> M = row0+v, lanes 16-31 -> M = row0+8+v; N = lane&15
  const int mb = row0 + ((lane >> 4) << 3);
#pragma unroll
  for (int v = 0; v < 8; ++v) {
    int r = mb + v;
    if (r < n_nodes) h1[(size_t)r * 16 + m] = c[v];
  }
}

// ---------------- layer 2 GEMM: z2 = hid @ W2  ([N,16]x[16,8], N-padded to 16) ----
__global__ void __launch_bounds__(256) gemm2_wmma(const float* __restrict__ hid,
                                                  const float* __restrict__ W2,
                                                  float* __restrict__ z2, int n_nodes) {
  __shared__ float sW[16 * 16];
  {
    int k = threadIdx.x >> 4, n = threadIdx.x & 15;  // 256 threads -> 16x16
    sW[k * 16 + n] = (n < 8) ? W2[k * 8 + n] : 0.0f;
  }
  __syncthreads();

  const int lane = threadIdx.x & 31;
  const int tile = blockIdx.x * 8 + (threadIdx.x >> 5);
  const int row0 = tile * 16;
  if (row0 >= n_nodes) return;

  const int m = lane & 15;
  const int koff = (lane >> 4) << 1;

  int rA = row0 + m;
  if (rA > n_nodes - 1) rA = n_nodes - 1;
  const float* __restrict__ hrow = hid + (size_t)rA * 16;

  v8f c = {0.f, 0.f, 0.f, 0.f, 0.f, 0.f, 0.f, 0.f};
#pragma unroll
  for (int k = 0; k < 16; k += 4) {
    v2f a, b;
    a[0] = hrow[k + koff];
    a[1] = hrow[k + koff + 1];
    b[0] = sW[(k + koff) * 16 + m];
    b[1] = sW[(k + koff + 1) * 16 + m];
    c = __builtin_amdgcn_wmma_f32_16x16x4_f32(false, a, false, b, (short)0, c,
                                              false, false);
  }

  const int mb = row0 + ((lane >> 4) << 3);
  if (m < 8) {  // only first 8 output columns are real
#pragma unroll
    for (int v = 0; v < 8; ++v) {
      int r = mb + v;
      if (r < n_nodes) z2[(size_t)r * 8 + m] = c[v];
    }
  }
}

// ---------------- edge aggregation (gather * norm, atomic scatter-add) ----------
// 16 lanes per edge: coalesced 64B gather of src features, L2-resident f32 atomics.
__global__ void aggregate16(const long long* __restrict__ rows,
                            const long long* __restrict__ cols,
                            const float* __restrict__ dinv,
                            const float* __restrict__ src, float* __restrict__ dst,
                            long long nE) {
  long long t = (long long)blockIdx.x * blockDim.x + threadIdx.x;
  long long e = t >> 4;
  if (e >= nE) return;
  int f = (int)(t & 15);
  long long r = rows[e], c = cols[e];
  float norm = dinv[r] * dinv[c];
  atomicAdd(&dst[c * 16 + f], src[r * 16 + f] * norm);
}

__global__ void aggregate8(const long long* __restrict__ rows,
                           const long long* __restrict__ cols,
                           const float* __restrict__ dinv,
                           const float* __restrict__ src, float* __restrict__ dst,
                           long long nE) {
  long long t = (long long)blockIdx.x * blockDim.x + threadIdx.x;
  long long e = t >> 3;
  if (e >= nE) return;
  int f = (int)(t & 7);
  long long r = rows[e], c = cols[e];
  float norm = dinv[r] * dinv[c];
  atomicAdd(&dst[c * 8 + f], src[r * 8 + f] * norm);
}

// ---------------- fused self-loop + bias + ReLU (in place: agg1 -> hidden) -------
__global__ void relu_bias_self(float* __restrict__ agg1, const float* __restrict__ h1,
                               const float* __restrict__ dinv,
                               const float* __restrict__ b1, int n_nodes) {
  long long t = (long long)blockIdx.x * blockDim.x + threadIdx.x;
  if (t >= (long long)n_nodes * 16) return;
  int i = (int)(t >> 4), f = (int)(t & 15);
  float di = dinv[i];
  float v = agg1[t] + h1[t] * di * di + b1[f];
  agg1[t] = fmaxf(v, 0.0f);
}

// ---------------- fused self-loop + bias + log_softmax -------------------------
__global__ void finalize(const float* __restrict__ agg2, const float* __restrict__ z2,
                         const float* __restrict__ dinv, const float* __restrict__ b2,
                         float* __restrict__ out, int n_nodes) {
  int i = blockIdx.x * blockDim.x + threadIdx.x;
  if (i >= n_nodes) return;
  float di = dinv[i];
  float sl = di * di;
  float v[8];
  float m = -3.4e38f;
#pragma unroll
  for (int f = 0; f < 8; ++f) {
    v[f] = agg2[(size_t)i * 8 + f] + z2[(size_t)i * 8 + f] * sl + b2[f];
    m = fmaxf(m, v[f]);
  }
  float s = 0.0f;
#pragma unroll
  for (int f = 0; f < 8; ++f) s += __expf(v[f] - m);
  float lse = __logf(s);
#pragma unroll
  for (int f = 0; f < 8; ++f) out[(size_t)i * 8 + f] = v[f] - m - lse;
}

// ---------------- launch --------------------------------------------------------

extern "C" void kernel_launch(void* const* d_in, const int* in_sizes, int n_in,
                              void* d_out, int out_size, void* d_ws, size_t ws_size,
                              hipStream_t stream) {
  const float* x = (const float*)d_in[0];
  const long long* ei = (const long long*)d_in[1];  // int64 edge_index [2,E]
  const float* W1 = (const float*)d_in[2];
  const float* b1 = (const float*)d_in[3];
  const float* W2 = (const float*)d_in[4];
  const float* b2 = (const float*)d_in[5];
  float* out = (float*)d_out;

  const int N = in_sizes[0] / 128;
  const long long E = (long long)in_sizes[1] / 2;
  const long long* rows = ei;      // sources
  const long long* cols = ei + E;  // targets

  // workspace layout (floats): deg/dinv | h1 | agg1(->hidden) | z2 | agg2
  float* ws = (float*)d_ws;
  float* deg  = ws;                       // N       (becomes dinv in place)
  float* h1   = deg + N;                  // N*16
  float* agg1 = h1 + (size_t)N * 16;      // N*16    (becomes hidden in place)
  float* z2   = agg1 + (size_t)N * 16;    // N*8
  float* agg2 = z2 + (size_t)N * 8;       // N*8
  (void)n_in; (void)out_size; (void)ws_size;

  const int B = 256;
  // init (workspace is poisoned; must re-init every call)
  fill_f32<<<(N + B - 1) / B, B, 0, stream>>>(deg, 1.0f, N);
  fill_f32<<<((long long)N * 16 + B - 1) / B, B, 0, stream>>>(agg1, 0.0f, (long long)N * 16);
  fill_f32<<<((long long)N * 8 + B - 1) / B, B, 0, stream>>>(agg2, 0.0f, (long long)N * 8);

  // symmetric normalization factors
  deg_count<<<(int)((E + B - 1) / B), B, 0, stream>>>(cols, deg, E);
  rsqrt_inplace<<<(N + B - 1) / B, B, 0, stream>>>(deg, N);

  const int tiles = (N + 15) / 16;
  const int gemmBlocks = (tiles + 7) / 8;  // 8 waves (wave32) per 256-thread block

  // layer 1
  gemm1_wmma<<<gemmBlocks, B, 0, stream>>>(x, W1, h1, N);
  aggregate16<<<(int)((E * 16 + B - 1) / B), B, 0, stream>>>(rows, cols, deg, h1, agg1, E);
  relu_bias_self<<<(int)(((long long)N * 16 + B - 1) / B), B, 0, stream>>>(agg1, h1, deg, b1, N);

  // layer 2
  gemm2_wmma<<<gemmBlocks, B, 0, stream>>>(agg1, W2, z2, N);
  aggregate8<<<(int)((E * 8 + B - 1) / B), B, 0, stream>>>(rows, cols, deg, z2, agg2, E);
  finalize<<<(N + B - 1) / B, B, 0, stream>>>(agg2, z2, deg, b2, out, N);
}